// learnable_mel_scale_filter_22814866276390
// MI455X (gfx1250) — compile-verified
//
#include <hip/hip_runtime.h>

typedef __attribute__((ext_vector_type(2))) float v2f;
typedef __attribute__((ext_vector_type(8))) float v8f;

#define N_FREQ   1024
#define N_CH     8
#define N_MELS   96
#define N_FILTER 16
#define BATCH    8
#define N_TIME   512
#define CS       (N_FREQ * N_TIME)   // channel stride in x, in floats (524288)

// Grid: (N_MELS, BATCH). Block: 128 threads = 4 wave32.
// Wave w computes 8 output tiles of 16 time-steps x 16 filters for mel band m,
// batch by, accumulating over K = 8*Lmax with V_WMMA_F32_16X16X4_F32.
extern "C" __global__ __launch_bounds__(128)
void mel_wmma_f32_kernel(const float* __restrict__ x,
                         const float* __restrict__ W,
                         const float* __restrict__ bias,
                         const int*   __restrict__ idx,
                         float* __restrict__ out,
                         int Lmax)
{
    extern __shared__ __align__(16) float smem[];
    const int K = 8 * Lmax;

    float* Wlds = smem;                               // [K][16] floats, k = l*8 + c
    int*   idxs = (int*)(Wlds + K * 16);              // [Lmax] freq indices
    float* tr   = (float*)(idxs + ((Lmax + 3) & ~3)); // 4 waves * 16 rows * 20 floats

    const int m   = blockIdx.x;
    const int by  = blockIdx.y;
    const int tid = threadIdx.x;

    // stage this band's frequency indices
    for (int l = tid; l < Lmax; l += 128)
        idxs[l] = idx[m * Lmax + l];

    // stage W[m] permuted: Wlds[(l*8+c)*16 + f] = W[m][f][c][l]
    const int tot = K * 16;
    for (int e = tid; e < tot; e += 128) {
        const int f = e & 15;
        const int k = e >> 4;
        const int c = k & 7;
        const int l = k >> 3;
        Wlds[e] = W[((size_t)(m * 16 + f) * 8 + c) * Lmax + l];
    }
    __syncthreads();

    const int lane = tid & 31;
    const int w    = tid >> 5;       // wave id 0..3
    const int row  = lane & 15;      // A row (time in tile) / B column (filter f)
    const int hi   = lane >> 4;      // half-wave: K-slots {0,1} vs {2,3}

    // accumulators pre-loaded with bias: every C VGPR of a lane is column f=row
    const float bb = bias[m * 16 + row];
    v8f acc[8];
#pragma unroll
    for (int p = 0; p < 8; ++p)
#pragma unroll
        for (int i = 0; i < 8; ++i)
            acc[p][i] = bb;

    const float* xb      = x + (size_t)by * N_CH * CS;     // x[by][0][0][0]
    const int    laneoff = 2 * hi * CS + w * 128 + row;    // per-lane invariant
    const float* wl      = Wlds + hi * 32 + row;           // B fragment base

    for (int l = 0; l < Lmax; ++l) {
        const int freq = idxs[l];
        const float* p0 = xb + (size_t)freq * N_TIME + laneoff; // channels 0..3
        const float* p4 = p0 + 4 * (size_t)CS;                  // channels 4..7
        const float* wp = wl + l * 128;

        // B fragments for k = l*8 + {0..3} and {4..7} (shared by all 8 tiles)
        v2f b0; b0.x = wp[0];  b0.y = wp[16];
        v2f b1; b1.x = wp[64]; b1.y = wp[80];

#pragma unroll
        for (int p = 0; p < 8; ++p) {
            v2f a0; a0.x = p0[p * 16]; a0.y = p0[p * 16 + CS];
            v2f a1; a1.x = p4[p * 16]; a1.y = p4[p * 16 + CS];
            acc[p] = __builtin_amdgcn_wmma_f32_16x16x4_f32(
                         false, a0, false, b0, (short)0, acc[p], false, false);
            acc[p] = __builtin_amdgcn_wmma_f32_16x16x4_f32(
                         false, a1, false, b1, (short)0, acc[p], false, false);
        }
    }

    // Epilogue: relu + per-wave LDS transpose -> coalesced b128 stores.
    // Fully unrolled so acc[] is never dynamically indexed (no scratch spill).
    // out layout: ((b*16 + f)*96 + m)*512 + t
    float*       trw = tr + w * (16 * 20);
    const size_t ob  = (size_t)by * (N_FILTER * N_MELS * N_TIME)
                     + (size_t)m * N_TIME;
#pragma unroll
    for (int p = 0; p < 8; ++p) {
#pragma unroll
        for (int i = 0; i < 8; ++i) {
            // C layout: lane, VGPR i -> (n_local = i + 8*hi, f = row)
            float v = acc[p][i];
            trw[row * 20 + hi * 8 + i] = v > 0.0f ? v : 0.0f;
        }
        const int t0 = w * 128 + p * 16;
#pragma unroll
        for (int j = 0; j < 2; ++j) {
            const int linear = lane + 32 * j;     // 64 float4 chunks / 32 lanes
            const int r      = linear >> 2;       // filter row 0..15
            const int cc     = linear & 3;        // 4-float chunk within row
            const float4 v   = *(const float4*)(trw + r * 20 + cc * 4);
            *(float4*)(out + ob + (size_t)r * (N_MELS * N_TIME) + t0 + cc * 4) = v;
        }
    }
}

extern "C" void kernel_launch(void* const* d_in, const int* in_sizes, int n_in,
                              void* d_out, int out_size, void* d_ws, size_t ws_size,
                              hipStream_t stream) {
    (void)n_in; (void)out_size; (void)d_ws; (void)ws_size;
    const float* x    = (const float*)d_in[0];
    const float* W    = (const float*)d_in[1];
    const float* bias = (const float*)d_in[2];
    const int*   idx  = (const int*)d_in[3];
    float*       out  = (float*)d_out;

    const int Lmax = in_sizes[1] / (N_MELS * N_FILTER * N_CH);
    const int K    = 8 * Lmax;

    const size_t shmem = (size_t)K * 16 * sizeof(float)          // Wlds
                       + (size_t)((Lmax + 3) & ~3) * sizeof(int) // idxs (padded)
                       + (size_t)4 * 16 * 20 * sizeof(float);    // transpose bufs

    dim3 grid(N_MELS, BATCH);
    mel_wmma_f32_kernel<<<grid, 128, shmem, stream>>>(x, W, bias, idx, out, Lmax);
}